// HodgkinHuxleyCoupledNeurons_15582141350010
// MI455X (gfx1250) — compile-verified
//
#include <hip/hip_runtime.h>

typedef __attribute__((ext_vector_type(2))) float v2f;
typedef __attribute__((ext_vector_type(4))) float v4f;
typedef __attribute__((ext_vector_type(8))) float v8f;

#define HH_B 256
#define HH_N 128
#define HH_E 512   // 4*N

// Workspace layout (floats):
//  [0,         32768)  Vex[B][N]      contiguous V
//  [32768,     49152)  gCoC[N][N]     g_C[i][j]/C[j]
//  [49152,     49280)  r[N]           row sums of gCoC
//  [49280,     49281)  S              total sum of gCoC
#define WS_V     0
#define WS_GCOC  32768
#define WS_R     49152
#define WS_S     49280

// ---------------- prep kernels (tiny) ----------------

__global__ void __launch_bounds__(256)
hh_prep_extractV(const float* __restrict__ y, float* __restrict__ Vex) {
  int t = blockIdx.x * 256 + threadIdx.x;        // 0..B*N-1
  int b = t >> 7, j = t & 127;
  Vex[t] = y[b * HH_E + 4 * j];
}

__global__ void __launch_bounds__(128)
hh_prep_gcoc(const float* __restrict__ gC, const float* __restrict__ C,
             float* __restrict__ gCoC, float* __restrict__ r) {
  __shared__ float red[HH_N];
  int i = blockIdx.x, j = threadIdx.x;
  float v = gC[i * HH_N + j] / C[j];
  gCoC[i * HH_N + j] = v;
  red[j] = v;
  __syncthreads();
  for (int s = HH_N / 2; s > 0; s >>= 1) {
    if (j < s) red[j] += red[j + s];
    __syncthreads();
  }
  if (j == 0) r[i] = red[0];
}

__global__ void __launch_bounds__(128)
hh_prep_total(const float* __restrict__ r, float* __restrict__ S) {
  __shared__ float red[HH_N];
  int j = threadIdx.x;
  red[j] = r[j];
  __syncthreads();
  for (int s = HH_N / 2; s > 0; s >>= 1) {
    if (j < s) red[j] += red[j + s];
    __syncthreads();
  }
  if (j == 0) *S = red[0];
}

// ---------------- WMMA coupling GEMM + ydot ----------------
// acc[b,i] = sum_j gCoC[i,j] * V[b,j]  via  D = A(16x4 of V) x B(4x16 of gCoC^T)
// 128 waves, one 16x16 (batch x neuron) tile each. K=128 in 4 chunks of 8
// wmma steps, register double-buffered so loads overlap the XDL pipe.

__global__ void __launch_bounds__(256)
hh_ydot_wmma(const float* __restrict__ y,   const float* __restrict__ Ic,
             const float* __restrict__ C,   const float* __restrict__ g_Na,
             const float* __restrict__ E_Na,const float* __restrict__ g_K,
             const float* __restrict__ E_K, const float* __restrict__ g_L,
             const float* __restrict__ E_L, const float* __restrict__ m_inf,
             const float* __restrict__ tau_m,const float* __restrict__ h_inf,
             const float* __restrict__ tau_h,const float* __restrict__ n_inf,
             const float* __restrict__ tau_n,
             const float* __restrict__ Vex, const float* __restrict__ gCoC,
             const float* __restrict__ r,   float* __restrict__ ydot) {
  const int lane = threadIdx.x & 31;
  const int wave = threadIdx.x >> 5;
  const int wid  = blockIdx.x * 8 + wave;        // 0..127
  const int tb   = (wid >> 3) << 4;              // batch-tile base (16 of 256)
  const int tn   = (wid & 7) << 4;               // neuron-tile base (8 of 128)
  const int lhi  = lane >> 4;                    // lane group 0/1
  const int llo  = lane & 15;

  // A: row M=llo of V-tile, K = k0 + 2*lhi + {0,1}  (ISA 16x4 f32 A layout)
  // B: col N=llo of output = row llo of gCoC, same K striping
  const float* Arow = Vex  + (size_t)(tb + llo) * HH_N + 2 * lhi;
  const float* Brow = gCoC + (size_t)(tn + llo) * HH_N + 2 * lhi;

  v2f abuf[2][8], bbuf[2][8];

  #pragma unroll
  for (int u = 0; u < 8; ++u) {
    abuf[0][u] = *(const v2f*)(Arow + 4 * u);
    bbuf[0][u] = *(const v2f*)(Brow + 4 * u);
  }

  v8f acc = {0.f, 0.f, 0.f, 0.f, 0.f, 0.f, 0.f, 0.f};
  #pragma unroll
  for (int c = 0; c < 4; ++c) {
    const int cur = c & 1, nxt = cur ^ 1;
    if (c < 3) {
      #pragma unroll
      for (int u = 0; u < 8; ++u) {
        abuf[nxt][u] = *(const v2f*)(Arow + 32 * (c + 1) + 4 * u);
        bbuf[nxt][u] = *(const v2f*)(Brow + 32 * (c + 1) + 4 * u);
      }
    }
    #pragma unroll
    for (int u = 0; u < 8; ++u) {
      acc = __builtin_amdgcn_wmma_f32_16x16x4_f32(
          /*neg_a=*/false, abuf[cur][u], /*neg_b=*/false, bbuf[cur][u],
          /*c_mod=*/(short)0, acc, /*reuse_a=*/false, /*reuse_b=*/false);
    }
  }

  // Epilogue: lane holds D[M = v + 8*lhi, N = llo] in acc[v]  (C/D f32 layout)
  const int i   = tn + llo;
  const float iC = 1.0f / C[i];
  const float gna = g_Na[i], ena = E_Na[i];
  const float gk  = g_K[i],  ek  = E_K[i];
  const float gl  = g_L[i],  el  = E_L[i];
  const float mi  = m_inf[i], tm = tau_m[i];
  const float hf  = h_inf[i], th = tau_h[i];
  const float nf  = n_inf[i], tnn = tau_n[i];
  const float ri  = r[i];

  #pragma unroll
  for (int v = 0; v < 8; ++v) {
    const int b = tb + v + 8 * lhi;
    v4f yv = *(const v4f*)(y + (size_t)b * HH_E + 4 * i);
    float V = yv.x, m = yv.y, h = yv.z, n = yv.w;
    float m3 = m * m * m;
    float n2 = n * n, n4 = n2 * n2;
    float Vdot = iC * (-gna * m3 * h * (V - ena)
                       - gk * n4 * (V - ek)
                       - gl * (V - el) + Ic[b])
                 + V * ri - acc[v];
    v4f o;
    o.x = Vdot;
    o.y = (mi - m) / tm;
    o.z = (hf - h) / th;
    o.w = (nf - n) / tnn;
    *(v4f*)(ydot + (size_t)b * HH_E + 4 * i) = o;
  }
}

// ---------------- J writer: the 268 MB streaming kernel ----------------
// Each thread owns one (row-within-batch, column-group) and writes it for 4
// consecutive batches, reusing the batch-invariant value across the 4 NT
// stores. s = row%4 is uniform per 128-thread half-block -> no divergence on
// the store path. J >> L2 (268 MB vs 192 MB) so stores are nontemporal B128.

__global__ void __launch_bounds__(256)
hh_jwrite(const float* __restrict__ y,
          const float* __restrict__ C,    const float* __restrict__ g_Na,
          const float* __restrict__ E_Na, const float* __restrict__ g_K,
          const float* __restrict__ E_K,  const float* __restrict__ g_L,
          const float* __restrict__ tau_m,const float* __restrict__ tau_h,
          const float* __restrict__ tau_n,
          const float* __restrict__ gCoC, const float* __restrict__ Sp,
          float* __restrict__ J) {
  const int half = threadIdx.x >> 7;               // 0..1
  const int j    = threadIdx.x & 127;              // column group
  const int p    = blockIdx.x;                     // 0..(256*64 - 1)
  const int rr   = ((p & 255) << 1) + half;        // row within batch, 0..511
  const int b0   = (p >> 8) << 2;                  // batch base: 0,4,...,252
  const int i    = rr >> 2;
  const int s    = rr & 3;

  if (s == 0) {
    const float base = -gCoC[i * HH_N + j];        // batch-invariant
    const float iC  = 1.0f / C[i];
    const float gna = g_Na[i], ena = E_Na[i];
    const float gk  = g_K[i],  ek  = E_K[i];
    const float gl  = g_L[i];
    const float Sv  = *Sp;
    #pragma unroll
    for (int t = 0; t < 4; ++t) {
      const int b = b0 + t;
      v4f o = {base, 0.f, 0.f, 0.f};
      if (j == i) {
        v4f yv = *(const v4f*)(y + (size_t)b * HH_E + 4 * i);
        float V = yv.x, m = yv.y, h = yv.z, n = yv.w;
        float m3 = m * m * m;
        float n2 = n * n, n4 = n2 * n2;
        o.x = base + iC * (-gl - gna * h * m3 - gk * n4) + Sv;
        float dNa = V - ena;
        o.y = -iC * (3.0f * gna * h * m * m * dNa);
        o.z = -iC * (gna * m3 * dNa);
        o.w = -iC * (4.0f * gk * n2 * n * (V - ek));
      }
      __builtin_nontemporal_store(
          o, (v4f*)(J + ((size_t)b * HH_E + rr) * HH_E + 4 * (size_t)j));
    }
  } else {
    v4f o = {0.f, 0.f, 0.f, 0.f};                  // batch-invariant row
    if (j == i) {
      if (s == 1)      o.y = -1.0f / tau_m[i];
      else if (s == 2) o.z = -1.0f / tau_h[i];
      else             o.w = -1.0f / tau_n[i];
    }
    #pragma unroll
    for (int t = 0; t < 4; ++t) {
      const int b = b0 + t;
      __builtin_nontemporal_store(
          o, (v4f*)(J + ((size_t)b * HH_E + rr) * HH_E + 4 * (size_t)j));
    }
  }
}

// ---------------- launch ----------------

extern "C" void kernel_launch(void* const* d_in, const int* in_sizes, int n_in,
                              void* d_out, int out_size, void* d_ws, size_t ws_size,
                              hipStream_t stream) {
  const float* y     = (const float*)d_in[0];
  const float* Ic    = (const float*)d_in[1];
  const float* C     = (const float*)d_in[2];
  const float* g_Na  = (const float*)d_in[3];
  const float* E_Na  = (const float*)d_in[4];
  const float* g_K   = (const float*)d_in[5];
  const float* E_K   = (const float*)d_in[6];
  const float* g_L   = (const float*)d_in[7];
  const float* E_L   = (const float*)d_in[8];
  const float* m_inf = (const float*)d_in[9];
  const float* tau_m = (const float*)d_in[10];
  const float* h_inf = (const float*)d_in[11];
  const float* tau_h = (const float*)d_in[12];
  const float* n_inf = (const float*)d_in[13];
  const float* tau_n = (const float*)d_in[14];
  const float* g_C   = (const float*)d_in[15];

  float* ws   = (float*)d_ws;
  float* Vex  = ws + WS_V;
  float* gCoC = ws + WS_GCOC;
  float* r    = ws + WS_R;
  float* S    = ws + WS_S;

  float* ydot = (float*)d_out;                      // B*512 floats
  float* J    = ydot + (size_t)HH_B * HH_E;         // B*512*512 floats

  hh_prep_extractV<<<(HH_B * HH_N) / 256, 256, 0, stream>>>(y, Vex);
  hh_prep_gcoc<<<HH_N, HH_N, 0, stream>>>(g_C, C, gCoC, r);
  hh_prep_total<<<1, HH_N, 0, stream>>>(r, S);

  hh_ydot_wmma<<<16, 256, 0, stream>>>(y, Ic, C, g_Na, E_Na, g_K, E_K, g_L, E_L,
                                       m_inf, tau_m, h_inf, tau_h, n_inf, tau_n,
                                       Vex, gCoC, r, ydot);

  // 256 row-pairs per batch-group, 64 batch-groups of 4
  hh_jwrite<<<256 * (HH_B / 4), 256, 0, stream>>>(y, C, g_Na, E_Na, g_K, E_K,
                                                  g_L, tau_m, tau_h, tau_n,
                                                  gCoC, S, J);
}